// STC_encoder_89919435309241
// MI455X (gfx1250) — compile-verified
//
#include <hip/hip_runtime.h>

typedef __attribute__((ext_vector_type(2))) float v2f;
typedef __attribute__((ext_vector_type(8))) float v8f;

#define FEAT 256
#define HID 128
#define RELS 2
#define KNBR 10
#define TILE_M 16
#define CW (FEAT + RELS * HID) // 512 combined width

__global__ __launch_bounds__(256) void stc_fused_kernel(
    const int* __restrict__ nodes,        // [B]
    const int* __restrict__ neigh_idx,    // [R][B][K]
    const float* __restrict__ features,   // [N][FEAT]
    const float* __restrict__ W_stc,      // [R][FEAT][HID]
    const float* __restrict__ W_det,      // [CW][HID]
    float* __restrict__ out,              // [B][HID]
    int B)
{
    __shared__ float lds_comb[TILE_M * CW];   // 16 x 512 = 32 KB
    __shared__ float lds_a[TILE_M * FEAT];    // 16 x 256 = 16 KB

    const int tid  = threadIdx.x;
    const int wave = tid >> 5;
    const int lane = tid & 31;
    const int half = lane >> 4;       // which half-wave (K-pair / M-offset select)
    const int lrow = lane & 15;       // matrix row/col within half-wave
    const int n0   = wave * 16;       // this wave's N tile (8 waves cover 128 cols)
    const int b0   = blockIdx.x * TILE_M;

    // ---- Phase 1: gather self features -> combined[:, 0:FEAT] -------------
    // thread == column; coalesced 1KB row reads
    for (int m = 0; m < TILE_M; ++m) {
        int b = b0 + m; if (b >= B) b = B - 1;
        const int src = nodes[b];
        lds_comb[m * CW + tid] = features[src * FEAT + tid];
    }

    // ---- Phase 2: per relation: gather-mean, then STC GEMM ----------------
    for (int r = 0; r < RELS; ++r) {
        // mean over K sampled neighbors into lds_a[16][256]
        for (int m = 0; m < TILE_M; ++m) {
            int b = b0 + m; if (b >= B) b = B - 1;
            const int* idx = &neigh_idx[(r * B + b) * KNBR];
            float acc = 0.0f;
            #pragma unroll
            for (int k = 0; k < KNBR; ++k) {
                acc += features[idx[k] * FEAT + tid];
            }
            lds_a[m * FEAT + tid] = acc * (1.0f / (float)KNBR);
        }
        __syncthreads();  // lds_a ready for all waves

        // [16 x 256] x [256 x 128] -> each wave computes a 16x16 tile at n0
        const float* __restrict__ Wr = W_stc + r * FEAT * HID;
        v8f c = {};
        #pragma unroll 8
        for (int k = 0; k < FEAT; k += 4) {
            const int kk = k + 2 * half;
            // A frag: row lrow, K = kk..kk+1 (8B-aligned LDS read -> ds_load_b64)
            v2f a = *(const v2f*)&lds_a[lrow * FEAT + kk];
            // B frag: W rows kk..kk+1, col n0+lrow (L2-resident weight reads)
            v2f bf;
            bf.x = Wr[(kk + 0) * HID + n0 + lrow];
            bf.y = Wr[(kk + 1) * HID + n0 + lrow];
            c = __builtin_amdgcn_wmma_f32_16x16x4_f32(
                    false, a, false, bf, (short)0, c, false, false);
        }

        // scatter D tile into combined[:, FEAT + r*HID + n0 .. +16]
        #pragma unroll
        for (int v = 0; v < 8; ++v) {
            const int mrow = v + 8 * half;
            lds_comb[mrow * CW + FEAT + r * HID + n0 + lrow] = c[v];
        }
        __syncthreads();  // combined cols done; lds_a reads finished before reuse
    }

    // ---- Phase 3: det GEMM [16 x 512] x [512 x 128], ReLU, store ----------
    v8f c = {};
    #pragma unroll 8
    for (int k = 0; k < CW; k += 4) {
        const int kk = k + 2 * half;
        v2f a = *(const v2f*)&lds_comb[lrow * CW + kk];
        v2f bf;
        bf.x = W_det[(kk + 0) * HID + n0 + lrow];
        bf.y = W_det[(kk + 1) * HID + n0 + lrow];
        c = __builtin_amdgcn_wmma_f32_16x16x4_f32(
                false, a, false, bf, (short)0, c, false, false);
    }

    #pragma unroll
    for (int v = 0; v < 8; ++v) {
        const int mrow = v + 8 * half;
        const int b = b0 + mrow;
        if (b < B) {
            const float x = c[v];
            out[b * HID + n0 + lrow] = x > 0.0f ? x : 0.0f;
        }
    }
}

extern "C" void kernel_launch(void* const* d_in, const int* in_sizes, int n_in,
                              void* d_out, int out_size, void* d_ws, size_t ws_size,
                              hipStream_t stream) {
    const int*   nodes     = (const int*)d_in[0];
    const int*   neigh_idx = (const int*)d_in[1];
    const float* features  = (const float*)d_in[2];
    const float* W_stc     = (const float*)d_in[3];
    const float* W_det     = (const float*)d_in[4];
    float*       out       = (float*)d_out;

    const int B = in_sizes[0];                 // 20000
    const int grid = (B + TILE_M - 1) / TILE_M; // 1250 blocks of 16 rows
    stc_fused_kernel<<<grid, 256, 0, stream>>>(nodes, neigh_idx, features,
                                               W_stc, W_det, out, B);
}